// GATLayer_27247272526392
// MI455X (gfx1250) — compile-verified
//
#include <hip/hip_runtime.h>
#include <math.h>

typedef float v2f __attribute__((ext_vector_type(2)));
typedef float v8f __attribute__((ext_vector_type(8)));

#define IN_F 128
#define HC   128
#define NHEAD 4
#define CHAN 32
#define NEG_SLOPE 0.2f
#define LDS_STRIDE 130  // 128 + 2 pad: avoids 64-bank conflicts on column reads

// ---- monotone float <-> uint encoding for atomic max over floats ----
__device__ __forceinline__ unsigned encF(float f) {
    unsigned b = __float_as_uint(f);
    return (b & 0x80000000u) ? ~b : (b | 0x80000000u);
}
__device__ __forceinline__ float decF(unsigned k) {
    unsigned b = (k & 0x80000000u) ? (k & 0x7FFFFFFFu) : ~k;
    return __uint_as_float(b);
}

// =====================================================================
// K1: h[N,128] = x[N,128] @ W[128,128] via V_WMMA_F32_16X16X4_F32.
// Block = 256 threads (8 waves). Block covers 32 rows; wave w owns column
// tile [16w,16w+16) for BOTH 16-row tiles (B regs reused across the two
// accumulators). A staged through LDS (padded), B streamed from global
// (W is 64KB -> L2 resident).
// =====================================================================
__global__ __launch_bounds__(256) void gat_gemm_wmma(
    const float* __restrict__ x, const float* __restrict__ W,
    float* __restrict__ h, int N)
{
    __shared__ float xs[32 * LDS_STRIDE];
    const int tid = threadIdx.x;
    const int rowBase = blockIdx.x * 32;

    // cooperative load of 32x128 x-strip into LDS
    for (int i = tid; i < 32 * IN_F; i += 256) {
        int r = i >> 7, c = i & 127;
        int rr = rowBase + r; if (rr >= N) rr = N - 1;   // clamp (harmless dup)
        xs[r * LDS_STRIDE + c] = x[(long long)rr * IN_F + c];
    }
    __syncthreads();

    const int wave = tid >> 5;
    const int lane = tid & 31;
    const int n0 = wave * 16;      // column tile base
    const int rt = lane & 15;      // row-in-tile (A) / col-in-tile (B,D)
    const int kh = lane >> 4;      // K-half select (lanes 16-31 hold K+2,K+3)

    v8f acc0 = {}; v8f acc1 = {};
    for (int k = 0; k < IN_F; k += 4) {
        const int kk = k + kh * 2;
        v2f b;
        b.x = W[(kk    ) * HC + n0 + rt];
        b.y = W[(kk + 1) * HC + n0 + rt];
        v2f a0 = *(const v2f*)&xs[(rt     ) * LDS_STRIDE + kk];   // rows 0-15
        v2f a1 = *(const v2f*)&xs[(16 + rt) * LDS_STRIDE + kk];   // rows 16-31
        acc0 = __builtin_amdgcn_wmma_f32_16x16x4_f32(
                   false, a0, false, b, (short)0, acc0, false, false);
        acc1 = __builtin_amdgcn_wmma_f32_16x16x4_f32(
                   false, a1, false, b, (short)0, acc1, false, false);
    }

    // D layout: VGPR v, lanes 0-15 -> M=v, lanes 16-31 -> M=v+8; N=lane&15
    #pragma unroll
    for (int v = 0; v < 8; ++v) {
        const int M = v + 8 * kh;
        int r0 = rowBase + M, r1 = rowBase + 16 + M;
        if (r0 < N) h[(long long)r0 * HC + n0 + rt] = acc0[v];
        if (r1 < N) h[(long long)r1 * HC + n0 + rt] = acc1[v];
    }
}

// =====================================================================
// K2: per-(node,head) attention scalars + init of max/denominator bufs
// =====================================================================
__global__ void gat_node_att(
    const float* __restrict__ h, const float* __restrict__ att_src,
    const float* __restrict__ att_dst, float* __restrict__ a_src,
    float* __restrict__ a_dst, unsigned* __restrict__ mEnc,
    float* __restrict__ denom, int N)
{
    int gid = blockIdx.x * blockDim.x + threadIdx.x;   // node*H + head
    if (gid >= N * NHEAD) return;
    int node = gid >> 2, head = gid & 3;
    const float* hp = h + (long long)node * HC + head * CHAN;
    const float* as = att_src + head * CHAN;
    const float* ad = att_dst + head * CHAN;
    float ss = 0.f, sd = 0.f;
    #pragma unroll
    for (int c = 0; c < CHAN; c += 4) {
        float4 hv = *(const float4*)(hp + c);
        float4 av = *(const float4*)(as + c);
        float4 dv = *(const float4*)(ad + c);
        ss += hv.x * av.x + hv.y * av.y + hv.z * av.z + hv.w * av.w;
        sd += hv.x * dv.x + hv.y * dv.y + hv.z * dv.z + hv.w * dv.w;
    }
    a_src[gid] = ss;
    a_dst[gid] = sd;
    mEnc[gid]  = 0x007FFFFFu;   // encF(-inf)
    denom[gid] = 0.f;
}

__global__ void gat_zero(float* __restrict__ p, long long n) {
    long long g = (long long)blockIdx.x * blockDim.x + threadIdx.x;
    if (g < n) p[g] = 0.f;
}

// =====================================================================
// K3: segment max over edges (+ self loops) via encoded uint atomicMax
// =====================================================================
__global__ void gat_edge_max(
    const long long* __restrict__ ei, const float* __restrict__ a_src,
    const float* __restrict__ a_dst, unsigned* __restrict__ mEnc,
    int E, int T)
{
    int t = blockIdx.x * blockDim.x + threadIdx.x;
    if (t >= T) return;
    long long s, d;
    if (t < E) { s = ei[t]; d = ei[(long long)E + t]; } else { s = d = t - E; }
    #pragma unroll
    for (int hh = 0; hh < NHEAD; ++hh) {
        float e = a_src[s * NHEAD + hh] + a_dst[d * NHEAD + hh];
        e = e > 0.f ? e : NEG_SLOPE * e;
        atomicMax(&mEnc[d * NHEAD + hh], encF(e));
    }
}

// =====================================================================
// K4: denom[dst,h] += exp(e - m[dst,h])
// =====================================================================
__global__ void gat_edge_sum(
    const long long* __restrict__ ei, const float* __restrict__ a_src,
    const float* __restrict__ a_dst, const unsigned* __restrict__ mEnc,
    float* __restrict__ denom, int E, int T)
{
    int t = blockIdx.x * blockDim.x + threadIdx.x;
    if (t >= T) return;
    long long s, d;
    if (t < E) { s = ei[t]; d = ei[(long long)E + t]; } else { s = d = t - E; }
    #pragma unroll
    for (int hh = 0; hh < NHEAD; ++hh) {
        float e = a_src[s * NHEAD + hh] + a_dst[d * NHEAD + hh];
        e = e > 0.f ? e : NEG_SLOPE * e;
        float ex = __expf(e - decF(mEnc[d * NHEAD + hh]));
        atomicAdd(&denom[d * NHEAD + hh], ex);
    }
}

// =====================================================================
// K5: weighted scatter. One wave32 per edge; lane owns 4 contiguous
// channels (head = lane>>3). float4 gather of h[src], 4 f32 atomics to out.
// =====================================================================
__global__ __launch_bounds__(256) void gat_scatter(
    const long long* __restrict__ ei, const float* __restrict__ a_src,
    const float* __restrict__ a_dst, const unsigned* __restrict__ mEnc,
    const float* __restrict__ denom, const float* __restrict__ h,
    float* __restrict__ out, int E, int T)
{
    long long gt = (long long)blockIdx.x * blockDim.x + threadIdx.x;
    int wid = (int)(gt >> 5);
    int lane = threadIdx.x & 31;
    if (wid >= T) return;
    long long s, d;
    if (wid < E) { s = ei[wid]; d = ei[(long long)E + wid]; } else { s = d = wid - E; }

    const int hh = lane >> 3;           // 8 lanes per head
    const int c0 = lane * 4;            // 4 contiguous channels per lane
    float e = a_src[s * NHEAD + hh] + a_dst[d * NHEAD + hh];
    e = e > 0.f ? e : NEG_SLOPE * e;
    float m = decF(mEnc[d * NHEAD + hh]);
    float alpha = __expf(e - m) / (denom[d * NHEAD + hh] + 1e-16f);

    float4 hv = *(const float4*)(h + s * HC + c0);
    float* op = out + d * HC + c0;
    atomicAdd(op + 0, alpha * hv.x);
    atomicAdd(op + 1, alpha * hv.y);
    atomicAdd(op + 2, alpha * hv.z);
    atomicAdd(op + 3, alpha * hv.w);
}

// =====================================================================
// K6: out = elu(out + bias), in place on d_out
// =====================================================================
__global__ void gat_finalize(float* __restrict__ out,
                             const float* __restrict__ bias, long long n)
{
    long long g = (long long)blockIdx.x * blockDim.x + threadIdx.x;
    if (g >= n) return;
    float v = out[g] + bias[g & (HC - 1)];
    out[g] = v > 0.f ? v : (__expf(v) - 1.f);
}

extern "C" void kernel_launch(void* const* d_in, const int* in_sizes, int n_in,
                              void* d_out, int out_size, void* d_ws, size_t ws_size,
                              hipStream_t stream)
{
    const float*     x       = (const float*)d_in[0];
    const long long* ei      = (const long long*)d_in[1];   // int64 [2,E]
    const float*     W       = (const float*)d_in[2];
    const float*     att_src = (const float*)d_in[3];
    const float*     att_dst = (const float*)d_in[4];
    const float*     bias    = (const float*)d_in[5];
    float*           out     = (float*)d_out;

    const int N = in_sizes[0] / IN_F;
    const int E = in_sizes[1] / 2;
    const int T = E + N;                 // edges + self loops

    // workspace carve-out
    char* w = (char*)d_ws;
    float*    h      = (float*)w;              w += (size_t)N * HC * sizeof(float);
    float*    a_src  = (float*)w;              w += (size_t)N * NHEAD * sizeof(float);
    float*    a_dst  = (float*)w;              w += (size_t)N * NHEAD * sizeof(float);
    unsigned* mEnc   = (unsigned*)w;           w += (size_t)N * NHEAD * sizeof(unsigned);
    float*    denom  = (float*)w;              /* w += ... */

    const int blk = 256;

    // K1: WMMA GEMM (N divisible by 32 for this problem; guards handle rest)
    gat_gemm_wmma<<<(N + 31) / 32, blk, 0, stream>>>(x, W, h, N);

    // K2: attention scalars + init
    gat_node_att<<<(N * NHEAD + blk - 1) / blk, blk, 0, stream>>>(
        h, att_src, att_dst, a_src, a_dst, mEnc, denom, N);

    // zero accumulator (d_out)
    long long nOut = (long long)N * HC;
    gat_zero<<<(unsigned)((nOut + blk - 1) / blk), blk, 0, stream>>>(out, nOut);

    // K3: segment max
    gat_edge_max<<<(T + blk - 1) / blk, blk, 0, stream>>>(ei, a_src, a_dst, mEnc, E, T);

    // K4: exp-sum denominators
    gat_edge_sum<<<(T + blk - 1) / blk, blk, 0, stream>>>(ei, a_src, a_dst, mEnc, denom, E, T);

    // K5: weighted scatter, wave per edge
    long long scatThreads = (long long)T * 32;
    gat_scatter<<<(unsigned)((scatThreads + blk - 1) / blk), blk, 0, stream>>>(
        ei, a_src, a_dst, mEnc, denom, h, out, E, T);

    // K6: bias + ELU
    gat_finalize<<<(unsigned)((nOut + blk - 1) / blk), blk, 0, stream>>>(out, bias, nOut);
}